// CorrelatedGRFS2_31129922961876
// MI455X (gfx1250) — compile-verified
//
#include <hip/hip_runtime.h>
#include <hip/hip_bf16.h>
#include <math.h>

// MI455X / gfx1250: wave32, WMMA. Bandwidth-dominated problem:
//   sqrt_covar = 624 MB, triangular half = 315 MB -> ~17-20 us floor @ 23.3 TB/s.
// Stage plan:
//   k0: zero-fill padded intermediates (t,g)
//   k1: DFT weight table + energy accumulators init; padded copy of pct
//   k2: complex channel mixing + mask + smooth + per-(b,i) energy partials (VALU, coalesced over m)
//   k3: ratio[bi] = sqrt(e0/e1)
//   k4: Legendre synthesis  (per-m 592x128x124 padded f32 GEMM, V_WMMA_F32_16X16X4_F32, guard-free loop)
//   k5: inverse rDFT as GEMM (35344x240x244 padded f32, V_WMMA_F32_16X16X4_F32, guard-free loop)
// Intermediates t/g (~35 MB each) stay L2-resident (192 MB L2).

typedef float v2f __attribute__((ext_vector_type(2)));
typedef float v8f __attribute__((ext_vector_type(8)));

#define C_CH  73
#define B_SZ  4
#define LMAXC 121
#define MMAXC 121
#define NLATC 121
#define NLONC 240
#define NBI   (B_SZ * C_CH)            // 292
#define NROW3 (B_SZ * C_CH * 2)        // 584   valid rows for Legendre GEMM (bi x re/im)
#define NROW3P 592                     // padded to 37 tiles of 16
#define LPAD  124                      // L padded to multiple of 4 (K dim of stage 4... stage 3)
#define NLATP 128                      // lat cols padded to 8 tiles of 16
#define NROW4 (B_SZ * C_CH * NLATC)    // 35332 valid rows for DFT GEMM (b,i,lat)
#define NROW4P 35344                   // padded to 2209 tiles of 16
#define KDFT  244                      // 2*MMAX padded to multiple of 4

// sizes (floats)
#define T_ELEMS   ((size_t)LPAD * MMAXC * NROW3P)   // 124*121*592  = 8,882,368
#define G_ELEMS   ((size_t)NROW4P * KDFT)           // 35344*244    = 8,623,936
#define W_ELEMS   ((size_t)KDFT * NLONC)            // 58,560
#define PCT_ELEMS ((size_t)LPAD * MMAXC * NLATP)    // 1,920,512

// ----------------------------------------------------------- zero fill (f4)
__global__ void grf_zero4(float4* __restrict__ p, long n4) {
    long i = (long)blockIdx.x * blockDim.x + threadIdx.x;
    long stride = (long)gridDim.x * blockDim.x;
    float4 z = make_float4(0.f, 0.f, 0.f, 0.f);
    for (; i < n4; i += stride) p[i] = z;
}

// ---------------------------------------------------------------- init tables
__global__ void grf_init(float* __restrict__ W, float* __restrict__ esums) {
    int idx = threadIdx.x + blockIdx.x * blockDim.x;
    if (idx < 2 * NBI) esums[idx] = 0.f;
    if (idx < KDFT * NLONC) {
        int k = idx / NLONC, n = idx % NLONC;
        float val = 0.f;
        if (k < 2 * MMAXC) {
            int mm = k >> 1, p = k & 1;
            float cm = (mm == 0 || mm == NLONC / 2) ? 1.f : 2.f;
            int ph = (mm * n) % NLONC;                       // exact phase reduction
            float ang = 6.28318530717958647692f * (float)ph / (float)NLONC;
            val = (p == 0) ? cm * cosf(ang) : -cm * sinf(ang);
        }
        W[idx] = val;                                        // rows 242..243 zero pad
    }
}

// ------------------------------------------- padded copy of pct: [124][121][128]
__global__ void grf_pctpad(const float* __restrict__ pct, float* __restrict__ pctP) {
    long idx = (long)blockIdx.x * blockDim.x + threadIdx.x;
    if (idx < (long)PCT_ELEMS) {
        int k = (int)(idx % NLATP);
        long lm = idx / NLATP;
        int m = (int)(lm % MMAXC);
        int l = (int)(lm / MMAXC);
        float v = 0.f;
        if (l < LMAXC && k < NLATC)
            v = pct[((size_t)l * MMAXC + m) * NLATC + k];
        pctP[idx] = v;
    }
}

// ------------------------------------------------- stage 1: mixing + energies
// grid: x = i-tile (19), y = l (121); block = 128 threads (one per m, padded)
__global__ void grf_mix(const float* __restrict__ noise, const float* __restrict__ cov,
                        const float* __restrict__ mean,  const float* __restrict__ smooth,
                        float* __restrict__ t, float* __restrict__ esums) {
    const int l  = blockIdx.y;
    const int i0 = blockIdx.x * 4;
    const int m  = threadIdx.x;
    const bool valid  = (m < MMAXC);
    const bool active = valid && (m <= l);                   // lower-triangular mask

    float accR[4][4], accI[4][4];                            // [i-sub][b]
#pragma unroll
    for (int a = 0; a < 4; ++a)
#pragma unroll
        for (int b = 0; b < 4; ++b) { accR[a][b] = 0.f; accI[a][b] = 0.f; }

    if (active) {
        const size_t lm = (size_t)l * MMAXC + m;
        for (int j = 0; j < C_CH; ++j) {
            float nr[4], ni[4];
#pragma unroll
            for (int b = 0; b < 4; ++b) {
                const float2 nv = *(const float2*)(noise + 2u * ((size_t)(b * C_CH + j) * (LMAXC * MMAXC) + lm));
                nr[b] = nv.x; ni[b] = nv.y;
            }
#pragma unroll
            for (int a = 0; a < 4; ++a) {
                const int i = i0 + a;
                if (i < C_CH) {                              // uniform over the wave
                    const float2 cv = *(const float2*)(cov + 2u * ((size_t)(i * C_CH + j) * (LMAXC * MMAXC) + lm));
#pragma unroll
                    for (int b = 0; b < 4; ++b) {
                        accR[a][b] = fmaf(cv.x, nr[b], fmaf(-cv.y, ni[b], accR[a][b]));
                        accI[a][b] = fmaf(cv.x, ni[b], fmaf( cv.y, nr[b], accI[a][b]));
                    }
                }
            }
        }
    }

    const float sm  = active ? smooth[l * MMAXC + m] : 0.f;
    const float sm2 = sm * sm;

#pragma unroll
    for (int a = 0; a < 4; ++a) {
        const int i = i0 + a;                                // uniform across wave
        float mr = 0.f, mi = 0.f;
        if (active && i < C_CH) {
            const float2 mv = *(const float2*)(mean + 2u * ((size_t)i * (LMAXC * MMAXC) + (size_t)l * MMAXC + m));
            mr = mv.x; mi = mv.y;
        }
#pragma unroll
        for (int b = 0; b < 4; ++b) {
            const float sr = active ? (accR[a][b] + mr) : 0.f;
            const float si = active ? (accI[a][b] + mi) : 0.f;
            float e0 = (i < C_CH) ? (sr * sr + si * si) : 0.f;
            float e1 = e0 * sm2;
            if (valid && i < C_CH) {
                float2* tp = (float2*)(t + ((size_t)l * MMAXC + m) * NROW3P + (size_t)(b * C_CH + i) * 2);
                *tp = make_float2(sr * sm, si * sm);         // zeros for m > l
            }
#pragma unroll                                               // wave32 tree reduction
            for (int off = 16; off > 0; off >>= 1) {
                e0 += __shfl_down(e0, off, 32);
                e1 += __shfl_down(e1, off, 32);
            }
            if ((threadIdx.x & 31) == 0 && i < C_CH) {
                atomicAdd(esums + (b * C_CH + i), e0);
                atomicAdd(esums + NBI + (b * C_CH + i), e1);
            }
        }
    }
}

// --------------------------------------------------- stage 2: energy ratio
__global__ void grf_ratio(const float* __restrict__ esums, float* __restrict__ ratio) {
    int i = threadIdx.x + blockIdx.x * blockDim.x;
    if (i < NBI) {
        float e0 = esums[i], e1 = esums[NBI + i];
        ratio[i] = (e1 > 0.f) ? sqrtf(e0 / e1) : 0.f;
    }
}

// ------------------------- stage 3: Legendre synthesis (WMMA f32 16x16x4)
// grid: x = nTile(8) * rowTile(37) = 296, y = m (121); block = 32 (one wave)
// all loads padded in-bounds -> guard-free inner loop
__global__ void grf_legendre(const float* __restrict__ t, const float* __restrict__ pctP,
                             float* __restrict__ g) {
    const int m  = blockIdx.y;
    const int nT = blockIdx.x & 7;
    const int rT = blockIdx.x >> 3;
    const int lane = threadIdx.x;
    const int lo = lane & 15, hi = lane >> 4;

    v8f acc = {};
    const int row  = rT * 16 + lo;                           // A: M striped over lanes 0-15 (<592)
    const int kcol = nT * 16 + lo;                           // B: N striped over lanes 0-15 (<128)
    const float* tA = t    + (size_t)m * NROW3P + row;       // + l*121*592 per step
    const float* pB = pctP + (size_t)m * NLATP  + kcol;      // + l*121*128 per step
#pragma unroll 4
    for (int kb = 0; kb < LPAD; kb += 4) {
        v2f a, b;
#pragma unroll
        for (int v = 0; v < 2; ++v) {
            const int ll = kb + 2 * hi + v;                  // K index per ISA x4-f32 layout
            a[v] = tA[(size_t)ll * (MMAXC * NROW3P)];
            b[v] = pB[(size_t)ll * (MMAXC * NLATP)];
        }
        acc = __builtin_amdgcn_wmma_f32_16x16x4_f32(false, a, false, b, (short)0, acc, false, false);
    }
#pragma unroll
    for (int r = 0; r < 8; ++r) {                            // D: vgpr r -> M = r + 8*hi
        const int row3 = rT * 16 + r + 8 * hi;
        if (row3 < NROW3 && kcol < NLATC) {
            const int bi = row3 >> 1, p = row3 & 1;
            g[((size_t)bi * NLATC + kcol) * KDFT + 2 * m + p] = acc[r];
        }
    }
}

// ----------------------------- stage 4: inverse rDFT GEMM (WMMA f32 16x16x4)
// grid: x = nTile(15), y = rowTile(2209); block = 32 (one wave); guard-free loop
__global__ void grf_idft(const float* __restrict__ g, const float* __restrict__ W,
                         const float* __restrict__ ratio, float* __restrict__ f) {
    const int nT = blockIdx.x;
    const int rT = blockIdx.y;
    const int lane = threadIdx.x;
    const int lo = lane & 15, hi = lane >> 4;

    v8f acc = {};
    const int row = rT * 16 + lo;                            // < 35344 (padded)
    const int n   = nT * 16 + lo;                            // 240 = 15*16, always valid
    const float* gA = g + (size_t)row * KDFT;
    const float* wB = W + n;
#pragma unroll 4
    for (int kb = 0; kb < KDFT; kb += 4) {
        v2f a, b;
#pragma unroll
        for (int v = 0; v < 2; ++v) {
            const int k = kb + 2 * hi + v;
            a[v] = gA[k];
            b[v] = wB[(size_t)k * NLONC];
        }
        acc = __builtin_amdgcn_wmma_f32_16x16x4_f32(false, a, false, b, (short)0, acc, false, false);
    }
#pragma unroll
    for (int r = 0; r < 8; ++r) {
        const int row4 = rT * 16 + r + 8 * hi;
        if (row4 < NROW4) {
            const int bi = row4 / NLATC;                     // fold energy ratio here
            f[(size_t)row4 * NLONC + n] = acc[r] * ratio[bi];
        }
    }
}

// -------------------------------------------------------------------- launch
extern "C" void kernel_launch(void* const* d_in, const int* in_sizes, int n_in,
                              void* d_out, int out_size, void* d_ws, size_t ws_size,
                              hipStream_t stream) {
    (void)in_sizes; (void)n_in; (void)out_size; (void)ws_size;
    const float* noise  = (const float*)d_in[0];
    const float* cov    = (const float*)d_in[1];
    const float* mean   = (const float*)d_in[2];
    const float* pct    = (const float*)d_in[3];
    const float* smooth = (const float*)d_in[4];
    // d_in[5] = mask: lower-triangular ones, hardcoded as (m <= l)
    float* f = (float*)d_out;

    // workspace layout (~78 MB total)
    float* ws    = (float*)d_ws;
    float* t     = ws;                       // padded spectrum [124 l][121 m][592 rows]
    float* g     = t + T_ELEMS;              // padded Legendre output [35344 rows][244 k]
    float* W     = g + G_ELEMS;              // DFT weights [244][240]
    float* pctP  = W + W_ELEMS;              // padded pct [124][121][128]
    float* esums = pctP + PCT_ELEMS;         // e0[292], e1[292]
    float* ratio = esums + 2 * NBI;          // [292]

    const long zero4 = (long)((T_ELEMS + G_ELEMS) / 4);      // both divisible by 4
    grf_zero4   <<<4096, 256, 0, stream>>>((float4*)t, zero4);
    grf_init    <<<(KDFT * NLONC + 255) / 256, 256, 0, stream>>>(W, esums);
    grf_pctpad  <<<(int)((PCT_ELEMS + 255) / 256), 256, 0, stream>>>(pct, pctP);
    grf_mix     <<<dim3((C_CH + 3) / 4, LMAXC), 128, 0, stream>>>(noise, cov, mean, smooth, t, esums);
    grf_ratio   <<<1, 320, 0, stream>>>(esums, ratio);
    grf_legendre<<<dim3(296, MMAXC), 32, 0, stream>>>(t, pctP, g);
    grf_idft    <<<dim3(15, NROW4P / 16), 32, 0, stream>>>(g, W, ratio, f);
}